// DDR_coarse_65326452572555
// MI455X (gfx1250) — compile-verified
//
#include <hip/hip_runtime.h>
#include <cstdint>
#include <cstddef>

typedef float v2f __attribute__((ext_vector_type(2)));
typedef float v8f __attribute__((ext_vector_type(8)));
typedef unsigned int u32x4 __attribute__((ext_vector_type(4)));
typedef int i32x4 __attribute__((ext_vector_type(4)));
typedef int i32x8 __attribute__((ext_vector_type(8)));

#define LEAKY(v) ((v) > 0.f ? (v) : 0.2f * (v))

// ---------------------------------------------------------------- utilities
__global__ __launch_bounds__(256) void k_fill0(float* __restrict__ p, int n) {
  int t = blockIdx.x * 256 + threadIdx.x;
  if (t < n) p[t] = 0.f;
}

__global__ __launch_bounds__(256) void k_concat2(const float* __restrict__ a, int ca,
                                                 const float* __restrict__ b, int cb,
                                                 float* __restrict__ y, int n) {
  int C = ca + cb;
  int t = blockIdx.x * 256 + threadIdx.x;
  if (t >= n * C) return;
  int r = t / C, c = t - r * C;
  y[t] = (c < ca) ? a[r * ca + c] : b[r * cb + (c - ca)];
}

// ------------------------------------------------- TDM 2D tile -> LDS (D#)
// Loads a (tile1 x tile0) tile of 4-byte elements from a row-major tensor
// (row stride = stride0 elements) into LDS at lds_off, row-major packed.
// tdim0/tdim1 are the remaining valid extents from the tile origin; the TDM
// zero-fills reads beyond them (free OOB padding).
__device__ __forceinline__ void tdm_load_2d(unsigned lds_off, const void* gptr,
                                            unsigned tdim0, unsigned tdim1,
                                            unsigned tile0, unsigned tile1,
                                            unsigned stride0) {
  unsigned long long ga = (unsigned long long)gptr;
  u32x4 g0;
  g0[0] = 1u;                                   // count=1, user descriptor
  g0[1] = lds_off;                              // lds_addr
  g0[2] = (unsigned)ga;                         // global_addr[31:0]
  g0[3] = (unsigned)(ga >> 32) | (2u << 30);    // global_addr[56:32] | type=2
  i32x8 g1;
  g1[0] = (int)(2u << 16);                      // data_size = 4 bytes
  g1[1] = (int)((tdim0 & 0xFFFFu) << 16);       // tensor_dim0[15:0]
  g1[2] = (int)((tdim0 >> 16) | ((tdim1 & 0xFFFFu) << 16));
  g1[3] = (int)((tdim1 >> 16) | (tile0 << 16)); // tile_dim0
  g1[4] = (int)(tile1 & 0xFFFFu);               // tile_dim1 (tile_dim2 = 0)
  g1[5] = (int)stride0;                         // tensor_dim0_stride[31:0]
  g1[6] = 0;
  g1[7] = 0;
  i32x4 z4 = {0, 0, 0, 0};                      // D# groups 2/3: dims 2-4 unused
  i32x8 z8 = {0, 0, 0, 0, 0, 0, 0, 0};
  __builtin_amdgcn_tensor_load_to_lds(g0, g1, z4, z4, z8, 0);
}

// ----------------------------------------------------------------- GEMM core
// Y[N,Cout] = A @ W (+bias).  GATHER==0: A row = X row.
// GATHER==1: one-ring gather, A[m,k] = X[hx[m*7 + (k>>5)]*32 + (k&31)]  (F=32).
// Block = 128 threads (4 waves) -> 64x16 output tile; K chunked by 32 through
// LDS; 8 x V_WMMA_F32_16X16X4_F32 per chunk. A/B tiles DMA'd by the TDM
// (tensor_load_to_lds, TENSORcnt), gather-A staged manually.
template <int GATHER>
__global__ __launch_bounds__(128) void k_gemm16t(
    const float* __restrict__ X, const float* __restrict__ W,
    const float* __restrict__ bias, float* __restrict__ Y,
    int N, int Cin, int Cout, const int* __restrict__ hx) {
  __shared__ float shA[64 * 32];
  __shared__ float shB[32 * 16];

  const int tid  = threadIdx.x;
  const int lane = tid & 31;
  const int wave = tid >> 5;
  const int col0 = blockIdx.y << 4;
  const int row0 = blockIdx.x << 6;

  const int h  = lane >> 4;   // K-pair select (A/B), row-half select (D)
  const int lm = lane & 15;   // A row-in-tile / B,D column
  v8f acc = {};

  const unsigned offA = (unsigned)__builtin_amdgcn_readfirstlane(
      (int)(unsigned long long)(uintptr_t)(void*)shA);
  const unsigned offB = (unsigned)__builtin_amdgcn_readfirstlane(
      (int)(unsigned long long)(uintptr_t)(void*)shB);

  const int nCh = (Cin + 31) >> 5;
  for (int ch = 0; ch < nCh; ++ch) {
    const int k0 = ch << 5;
    __syncthreads();  // previous chunk's LDS reads must finish before DMA
    if (tid == 0) {
      // B panel: 32 x 16 from W[k0.., col0..]
      tdm_load_2d(offB, W + k0 * Cout + col0,
                  (unsigned)(Cout - col0), (unsigned)(Cin - k0),
                  16u, 32u, (unsigned)Cout);
      if (!GATHER) {
        // A tile: 64 rows x 32 k from X[row0.., k0..]
        tdm_load_2d(offA, X + row0 * Cin + k0,
                    (unsigned)(Cin - k0), (unsigned)(N - row0),
                    32u, 64u, (unsigned)Cin);
      }
    }
    if (GATHER) {
      for (int idx = tid; idx < 64 * 32; idx += 128) {
        int r = idx >> 5, kk = idx & 31;
        int row = row0 + r, k = k0 + kk;
        float v = 0.f;
        if (row < N && k < Cin) {
          int i = k >> 5, c = k & 31;  // one-ring feature width F == 32
          v = X[hx[row * 7 + i] * 32 + c];
        }
        shA[idx] = v;
      }
    }
    if (tid == 0) __builtin_amdgcn_s_wait_tensorcnt(0);
    __syncthreads();

    const float* aRow = shA + ((wave << 4) + lm) * 32;
#pragma unroll
    for (int kk = 0; kk < 32; kk += 4) {
      int ka = kk + 2 * h;
      v2f a, b;
      a.x = aRow[ka];
      a.y = aRow[ka + 1];
      b.x = shB[ka * 16 + lm];
      b.y = shB[(ka + 1) * 16 + lm];
      acc = __builtin_amdgcn_wmma_f32_16x16x4_f32(false, a, false, b, (short)0,
                                                  acc, false, false);
    }
  }

  const int rw0 = row0 + (wave << 4);
  const int n = col0 + lm;
  if (n < Cout) {
    float bv = bias ? bias[n] : 0.f;
#pragma unroll
    for (int r = 0; r < 8; ++r) {
      int row = rw0 + r + 8 * h;
      if (row < N) Y[row * Cout + n] = acc[r] + bv;
    }
  }
}

// -------------------------------------------------------------- GMM pieces
__global__ __launch_bounds__(256) void k_edgew(const float* __restrict__ ps,
                                               const float* __restrict__ mu,
                                               const float* __restrict__ sg,
                                               float* __restrict__ w, int E) {
  int e = blockIdx.x * 256 + threadIdx.x;
  if (e >= E) return;
  float p0 = ps[2 * e], p1 = ps[2 * e + 1];
#pragma unroll
  for (int k = 0; k < 3; ++k) {
    float d0 = p0 - mu[2 * k], d1 = p1 - mu[2 * k + 1];
    float s0 = sg[2 * k], s1 = sg[2 * k + 1];
    float q = d0 * d0 / (s0 * s0 + 1e-15f) + d1 * d1 / (s1 * s1 + 1e-15f);
    w[3 * e + k] = expf(-0.5f * q);
  }
}

__global__ __launch_bounds__(256) void k_scatter(
    const float* __restrict__ XG, const int* __restrict__ src,
    const int* __restrict__ dst, const float* __restrict__ w,
    float* __restrict__ S, float* __restrict__ cnt, int E, int Cout) {
  int t = blockIdx.x * 256 + threadIdx.x;
  if (t >= E * Cout) return;
  int e = t / Cout, c = t - e * Cout;
  int s = src[e], d = dst[e];
  const float* xr = XG + s * (3 * Cout);
  __builtin_prefetch(xr + c, 0, 0);  // global_prefetch_b8
  float m = w[3 * e] * xr[c] + w[3 * e + 1] * xr[Cout + c] +
            w[3 * e + 2] * xr[2 * Cout + c];
  atomicAdd(S + d * Cout + c, m);
  if (c == 0) atomicAdd(cnt + d, 1.f);
}

__global__ __launch_bounds__(256) void k_finalize(
    const float* __restrict__ S, const float* __restrict__ cnt,
    const float* __restrict__ XR, const float* __restrict__ bias,
    float* __restrict__ Y, int N, int C) {
  int t = blockIdx.x * 256 + threadIdx.x;
  if (t >= N * C) return;
  int r = t / C, c = t - r * C;
  float cv = cnt[r];
  cv = cv > 1.f ? cv : 1.f;
  float v = S[t] / cv + XR[t] + bias[c];
  Y[t] = LEAKY(v);
}

// torch-faithful pool: y[n,f] = mean_j flat[f*7+j], flat[p]=x[hx[n,p/F], p%F]
__global__ __launch_bounds__(256) void k_hexpool(
    const float* __restrict__ x, const int* __restrict__ hx,
    float* __restrict__ y, int nOut, int F) {
  int t = blockIdx.x * 256 + threadIdx.x;
  if (t >= nOut * F) return;
  int n = t / F, f = t - n * F;
  float s = 0.f;
  int base = f * 7;
#pragma unroll
  for (int j = 0; j < 7; ++j) {
    int p2 = base + j;
    int i = p2 / F, c = p2 - i * F;
    s += x[hx[n * 7 + i] * F + c];
  }
  y[t] = s * (1.f / 7.f);
}

__global__ __launch_bounds__(256) void k_hexup(
    const float* __restrict__ x, const int* __restrict__ up,
    float* __restrict__ y, int nOld, int nNew, int F) {
  int t = blockIdx.x * 256 + threadIdx.x;
  if (t >= nNew * F) return;
  int r = t / F, f = t - r * F;
  if (r < nOld) {
    y[t] = x[r * F + f];
  } else {
    int q = r - nOld;
    int u0 = up[2 * q], u1 = up[2 * q + 1];
    y[t] = 0.5f * (x[u0 * F + f] + x[u1 * F + f]);
  }
}

__global__ __launch_bounds__(32) void k_softmax162(float* __restrict__ x, int n, int C) {
  int row = blockIdx.x;
  if (row >= n) return;
  float* r = x + (size_t)row * C;
  int lane = threadIdx.x;
  float mx = -1e30f;
  for (int c = lane; c < C; c += 32) mx = fmaxf(mx, r[c]);
  for (int o = 16; o; o >>= 1) mx = fmaxf(mx, __shfl_xor(mx, o, 32));
  float sm = 0.f;
  for (int c = lane; c < C; c += 32) sm += expf(r[c] - mx);
  for (int o = 16; o; o >>= 1) sm += __shfl_xor(sm, o, 32);
  float inv = 1.f / sm;
  for (int c = lane; c < C; c += 32) r[c] = expf(r[c] - mx) * inv;
}

static inline unsigned cdiv(long long a, long long b) { return (unsigned)((a + b - 1) / b); }

extern "C" void kernel_launch(void* const* d_in, const int* in_sizes, int n_in,
                              void* d_out, int out_size, void* d_ws, size_t ws_size,
                              hipStream_t stream) {
  static const int LEV[6] = {40962, 10242, 2562, 642, 162, 42};
  static const int EDG[6] = {245760, 61440, 15360, 3840, 960, 240};
  static const int CI[22] = {8, 32, 32, 64, 64, 128, 128, 256, 256, 512, 512, 512,
                             768, 256, 384, 128, 192, 64, 96, 32, 40, 32};
  static const int CO[22] = {32, 32, 64, 64, 128, 128, 256, 256, 512, 512, 512, 512,
                             256, 256, 128, 128, 64, 64, 32, 32, 32, 32};
  (void)in_sizes; (void)n_in; (void)out_size; (void)ws_size;

  int p = 0;
  const float* moving = (const float*)d_in[p++];
  const float* target = (const float*)d_in[p++];
  p++;  // edge_input == edge_indexes[0]
  const int* hexes[6];
  for (int i = 0; i < 6; ++i) hexes[i] = (const int*)d_in[p++];
  const int* eidx[6];
  for (int i = 0; i < 6; ++i) eidx[i] = (const int*)d_in[p++];
  const float* pseudo[6];
  for (int i = 0; i < 6; ++i) pseudo[i] = (const float*)d_in[p++];
  const int* ups[6];
  for (int i = 0; i < 6; ++i) ups[i] = (const int*)d_in[p++];
  struct GP { const float *g, *mu, *sigma, *root, *bias; } gp[22];
  for (int i = 0; i < 22; ++i) {
    gp[i].g = (const float*)d_in[p++];
    gp[i].mu = (const float*)d_in[p++];
    gp[i].sigma = (const float*)d_in[p++];
    gp[i].root = (const float*)d_in[p++];
    gp[i].bias = (const float*)d_in[p++];
  }
  const float* labW = (const float*)d_in[p++];
  const float* labBias = (const float*)d_in[p++];

  char* wsp = (char*)d_ws;
  auto alloc = [&](long long nf) -> float* {
    float* r = (float*)wsp;
    wsp += ((nf * 4 + 255) / 256) * 256;
    return r;
  };
  float* xin   = alloc(40962LL * 8);
  float* skip1 = alloc(10242LL * 32);
  float* skip2 = alloc(2562LL * 64);
  float* skip3 = alloc(642LL * 128);
  float* skip4 = alloc(162LL * 256);
  float* s5    = alloc(42LL * 512);
  float* bufA  = alloc(40962LL * 32);
  float* bufB  = alloc(40962LL * 32);
  float* bufU  = alloc(40962LL * 32);
  float* bufC  = alloc(40962LL * 40);
  float* XG    = alloc(40962LL * 96);
  float* XR    = alloc(40962LL * 32);
  float* S     = alloc(40962LL * 32);
  float* cnt   = alloc(40962LL);
  float* wE    = alloc(245760LL * 3);
  float* labA  = alloc(40962LL * 162);
  float* labPB = alloc(10242LL * 162);
  const float* skipPtr[5] = {xin, skip1, skip2, skip3, skip4};
  const int skipF[5] = {8, 32, 64, 128, 256};

  auto fill0 = [&](float* ptr, long long n) {
    k_fill0<<<cdiv(n, 256), 256, 0, stream>>>(ptr, (int)n);
  };
  auto gemm = [&](const float* Xp, const float* Wp, const float* bp, float* Yp,
                  int N, int Cin, int Cout, const int* hx) {
    dim3 grid(cdiv(N, 64), cdiv(Cout, 16));
    if (hx) k_gemm16t<1><<<grid, 128, 0, stream>>>(Xp, Wp, bp, Yp, N, Cin, Cout, hx);
    else    k_gemm16t<0><<<grid, 128, 0, stream>>>(Xp, Wp, bp, Yp, N, Cin, Cout, nullptr);
  };
  auto run_gmm = [&](int ci, const float* Xp, float* Yp, int li) {
    int N = LEV[li], E = EDG[li], Cin = CI[ci], Cout = CO[ci];
    gemm(Xp, gp[ci].g, nullptr, XG, N, Cin, 3 * Cout, nullptr);
    gemm(Xp, gp[ci].root, nullptr, XR, N, Cin, Cout, nullptr);
    k_edgew<<<cdiv(E, 256), 256, 0, stream>>>(pseudo[li], gp[ci].mu, gp[ci].sigma, wE, E);
    fill0(S, (long long)N * Cout);
    fill0(cnt, N);
    k_scatter<<<cdiv((long long)E * Cout, 256), 256, 0, stream>>>(
        XG, eidx[li], eidx[li] + E, wE, S, cnt, E, Cout);
    k_finalize<<<cdiv((long long)N * Cout, 256), 256, 0, stream>>>(
        S, cnt, XR, gp[ci].bias, Yp, N, Cout);
  };

  // x_in = concat(moving, target)
  k_concat2<<<cdiv(40962LL * 8, 256), 256, 0, stream>>>(moving, 4, target, 4, xin, 40962);

  // encoder
  const float* cur = xin;
  for (int i = 0; i < 5; ++i) {
    run_gmm(2 * i, cur, bufA, i);
    run_gmm(2 * i + 1, bufA, bufB, i);
    float* dstP = (i < 4) ? (float*)skipPtr[i + 1] : s5;
    int nOut = LEV[i + 1], F = CO[2 * i + 1];
    k_hexpool<<<cdiv((long long)nOut * F, 256), 256, 0, stream>>>(bufB, hexes[i], dstP, nOut, F);
    cur = dstP;
  }
  // bottleneck (ico1)
  run_gmm(10, cur, bufA, 5);
  run_gmm(11, bufA, bufB, 5);
  cur = bufB;
  int curF = 512;
  // decoder
  int ci = 12;
  for (int i = 4; i >= 0; --i) {
    int nNew = LEV[i], nOld = LEV[i + 1];
    k_hexup<<<cdiv((long long)nNew * curF, 256), 256, 0, stream>>>(cur, ups[i], bufU, nOld, nNew, curF);
    k_concat2<<<cdiv((long long)nNew * (curF + skipF[i]), 256), 256, 0, stream>>>(
        bufU, curF, skipPtr[i], skipF[i], bufC, nNew);
    run_gmm(ci, bufC, bufA, i);
    run_gmm(ci + 1, bufA, bufB, i);
    curF = CO[ci + 1];
    ci += 2;
    cur = bufB;
  }

  // label head: gathered one-ring GEMM (7*32 -> 162) + softmax + 4 pools
  gemm(cur, labW, labBias, labA, 40962, 224, 162, hexes[0]);
  k_softmax162<<<40962, 32, 0, stream>>>(labA, 40962, 162);
  k_hexpool<<<cdiv(10242LL * 162, 256), 256, 0, stream>>>(labA, hexes[0], labPB, 10242, 162);
  k_hexpool<<<cdiv(2562LL * 162, 256), 256, 0, stream>>>(labPB, hexes[1], labA, 2562, 162);
  k_hexpool<<<cdiv(642LL * 162, 256), 256, 0, stream>>>(labA, hexes[2], labPB, 642, 162);
  k_hexpool<<<cdiv(162LL * 162, 256), 256, 0, stream>>>(labPB, hexes[3], (float*)d_out, 162, 162);
}